// Flow3D_6081673691378
// MI455X (gfx1250) — compile-verified
//
#include <hip/hip_runtime.h>

typedef float v2f __attribute__((ext_vector_type(2)));
typedef float v8f __attribute__((ext_vector_type(8)));

#define B_  8
#define K_  8
#define H_  384
#define W_  384
#define HW_ (H_*W_)

// ---- pass 1: zero the 256 accumulator floats --------------------------------
__global__ void zero_sums(float* __restrict__ s) {
    s[threadIdx.x] = 0.0f;   // exactly 256 floats: B*K*4
}

// ---- pass 1: centroid reduction as a skinny GEMM via V_WMMA_F32_16X16X4_F32 -
// S[b,k,c] = sum_pix mask[b,k,pix] * comp(c,pix),  comp = (wx, wy, wz, 1)
// A(16x4): rows = k (8 used), K-dim = 4 consecutive pixels
// B(4x16): rows = pixels,     cols  = world component (4 used)
#define RBLK 72              // blocks per batch
#define RCHUNK (HW_/RBLK)    // 2048 pixels per block
#define RWAVE  (RCHUNK/8)    // 256 pixels per wave
#define RITERS (RWAVE/4)     // 64 WMMA steps per wave

__global__ void __launch_bounds__(256)
reduce_wmma(const float* __restrict__ mask,
            const float* __restrict__ depth,
            float* __restrict__ sums) {
    const int b      = blockIdx.y;
    const int waveId = threadIdx.x >> 5;
    const int lane   = threadIdx.x & 31;
    const int idx    = lane & 15;        // A row (k) and B col (component)
    const int h2     = lane >> 4;        // lane half -> K slots {2h, 2h+1}

    const int  kk    = (idx < 8) ? idx : 7;            // clamped, value zeroed below
    const bool isA   = (idx < 8);
    const float sx = (idx == 0) ? 1.f : 0.f;           // one-hot component selectors
    const float sy = (idx == 1) ? 1.f : 0.f;
    const float sd = (idx == 2) ? 1.f : 0.f;
    const float s1 = (idx == 3) ? 1.f : 0.f;

    const int waveBase = blockIdx.x * RCHUNK + waveId * RWAVE;
    const int start    = waveBase + 2*h2;              // lane's first pixel (always even)

    const float* __restrict__ mp = mask  + (size_t)(b*K_ + kk) * HW_ + start;
    const float* __restrict__ dp = depth + (size_t)b * HW_ + start;

    // pixel-coordinate recurrences: px = col*(2/W) + (1/W - 1), py likewise.
    // col is always even and steps by +4, so col+1 never wraps a row.
    const float kxs = 2.0f / (float)W_;
    const float kys = 2.0f / (float)H_;
    int col = start % W_;
    int row = start / W_;                              // one-time integer div
    float pxc = (float)col * kxs + (1.0f/(float)W_ - 1.0f);
    float pyc = (float)row * kys + (1.0f/(float)H_ - 1.0f);
    const float sxkx = sx * kxs;                       // px(p0+1) = px(p0) + 2/W

    v8f acc0 = {}, acc1 = {};
    #pragma unroll 4
    for (int it = 0; it < RITERS; ++it) {
        if ((it & 15) == 0) {                          // wave-uniform scalar branch
            __builtin_prefetch(mp + 512, 0, 0);        // global_prefetch_b8, ~2KB ahead
            __builtin_prefetch(dp + 512, 0, 0);
        }
        // A: contiguous mask pair (b64), rows 8..15 forced to zero
        float m0 = mp[0], m1 = mp[1];
        v2f a;  a.x = isA ? m0 : 0.f;  a.y = isA ? m1 : 0.f;
        // B: world components of the same two pixels, branchless one-hot mix
        float d0 = dp[0], d1 = dp[1];
        float w0 = fmaf(sx, pxc, fmaf(sy, pyc, sd));   // comp weight at p0
        float w1 = w0 + sxkx;                          // comp weight at p0+1
        v2f bv;
        bv.x = fmaf(d0, w0, s1);
        bv.y = fmaf(d1, w1, s1);
        // D(16x16,f32) += A(16x4) x B(4x16); two accumulators for ILP
        if (it & 1)
            acc1 = __builtin_amdgcn_wmma_f32_16x16x4_f32(
                       false, a, false, bv, (short)0, acc1, false, false);
        else
            acc0 = __builtin_amdgcn_wmma_f32_16x16x4_f32(
                       false, a, false, bv, (short)0, acc0, false, false);
        // advance 4 pixels with branchless row wrap (W % 4 == 0)
        mp += 4; dp += 4;
        col += 4;
        bool wrap = (col >= W_);
        col = wrap ? col - W_ : col;
        row += wrap ? 1 : 0;
        pxc = wrap ? ((float)col * kxs + (1.0f/(float)W_ - 1.0f)) : (pxc + 4.0f*kxs);
        pyc = wrap ? (pyc + kys) : pyc;
    }
    v8f acc = acc0 + acc1;

    // C/D layout: VGPR j -> row M=j on lanes 0-15; lane's column N = idx.
    if (h2 == 0 && idx < 4) {
        #pragma unroll
        for (int j = 0; j < 8; ++j)
            atomicAdd(&sums[(b*K_ + j)*4 + idx], acc[j]);
    }
}

// ---- pass 2: fold everything into 108 constants per batch -------------------
__device__ inline void euler2mat(float ax, float ay, float az, float R[9]) {
    float cx = cosf(ax), sxn = sinf(ax);
    float cy = cosf(ay), syn = sinf(ay);
    float cz = cosf(az), szn = sinf(az);
    R[0] = cy*cz;                 R[1] = -cy*szn;               R[2] = syn;
    R[3] = cx*szn + sxn*syn*cz;   R[4] = cx*cz - sxn*syn*szn;   R[5] = -sxn*cy;
    R[6] = sxn*szn - cx*syn*cz;   R[7] = sxn*cz + cx*syn*szn;   R[8] = cx*cy;
}

__global__ void make_consts(const float* __restrict__ sums,
                            const float* __restrict__ T,
                            const float* __restrict__ Tc,
                            float* __restrict__ cst) {
    int t = threadIdx.x;
    if (t >= B_*K_) return;
    int b = t >> 3, k = t & 7;

    const float* s = sums + (b*K_ + k)*4;
    float tm = s[3] + 1e-6f;
    float p0 = s[0]/tm, p1 = s[1]/tm, p2 = s[2]/tm;

    const float* Tk = T + (b*K_ + k)*6;
    float Rk[9]; euler2mat(Tk[3], Tk[4], Tk[5], Rk);
    const float* Tcb = Tc + b*9;
    float Rc[9]; euler2mat(Tcb[3], Tcb[4], Tcb[5], Rc);

    // c_k = p + t - Rk@p
    float ck0 = p0 + Tk[0] - (Rk[0]*p0 + Rk[1]*p1 + Rk[2]*p2);
    float ck1 = p1 + Tk[1] - (Rk[3]*p0 + Rk[4]*p1 + Rk[5]*p2);
    float ck2 = p2 + Tk[2] - (Rk[6]*p0 + Rk[7]*p1 + Rk[8]*p2);

    // A = Rc @ (Rk - I)
    float M[9];
    #pragma unroll
    for (int i = 0; i < 9; ++i) M[i] = Rk[i];
    M[0] -= 1.f; M[4] -= 1.f; M[8] -= 1.f;
    float A[9];
    #pragma unroll
    for (int i = 0; i < 3; ++i)
        #pragma unroll
        for (int j = 0; j < 3; ++j)
            A[i*3+j] = Rc[i*3+0]*M[0*3+j] + Rc[i*3+1]*M[1*3+j] + Rc[i*3+2]*M[2*3+j];
    float bx = Rc[0]*ck0 + Rc[1]*ck1 + Rc[2]*ck2;
    float by = Rc[3]*ck0 + Rc[4]*ck1 + Rc[5]*ck2;
    float bz = Rc[6]*ck0 + Rc[7]*ck1 + Rc[8]*ck2;

    float* base = cst + b*128;
    if (k == 0) {
        #pragma unroll
        for (int i = 0; i < 9; ++i) base[i] = Rc[i];
        float pcx = Tcb[6], pcy = Tcb[7], pcz = Tcb[8];
        base[ 9] = pcx + Tcb[0] - (Rc[0]*pcx + Rc[1]*pcy + Rc[2]*pcz);
        base[10] = pcy + Tcb[1] - (Rc[3]*pcx + Rc[4]*pcy + Rc[5]*pcz);
        base[11] = pcz + Tcb[2] - (Rc[6]*pcx + Rc[7]*pcy + Rc[8]*pcz);
    }
    float* ak = base + 12 + k*12;
    #pragma unroll
    for (int i = 0; i < 9; ++i) ak[i] = A[i];
    ak[ 9] = bx; ak[10] = by; ak[11] = bz;
}

// ---- pass 3: per-pixel flow -------------------------------------------------
__global__ void __launch_bounds__(256)
flow_main(const float* __restrict__ mask, const float* __restrict__ depth,
          const float* __restrict__ cst, float* __restrict__ out) {
    __shared__ float C[108];
    const int b = blockIdx.y;
    if (threadIdx.x < 108) C[threadIdx.x] = cst[b*128 + threadIdx.x];
    __syncthreads();

    const int pix = blockIdx.x*256 + threadIdx.x;
    const float d = depth[(size_t)b*HW_ + pix];
    const int col = pix % W_, row = pix / W_;
    const float px = fmaf((float)col, 2.0f/(float)W_, 1.0f/(float)W_ - 1.0f);
    const float py = fmaf((float)row, 2.0f/(float)H_, 1.0f/(float)H_ - 1.0f);
    const float wx = d*px, wy = d*py, wz = d;

    float vx = C[0]*wx + C[1]*wy + C[2]*wz + C[ 9];
    float vy = C[3]*wx + C[4]*wy + C[5]*wz + C[10];
    float vz = C[6]*wx + C[7]*wy + C[8]*wz + C[11];

    const float* __restrict__ mp = mask + (size_t)b*K_*HW_ + pix;
    #pragma unroll
    for (int k = 0; k < K_; ++k) {
        float m = mp[(size_t)k*HW_];
        const float* a = &C[12 + k*12];
        vx += m*(a[0]*wx + a[1]*wy + a[2]*wz + a[ 9]);
        vy += m*(a[3]*wx + a[4]*wy + a[5]*wz + a[10]);
        vz += m*(a[6]*wx + a[7]*wy + a[8]*wz + a[11]);
    }
    const float inv = 1.0f / vz;
    float2 f; f.x = vx*inv - px; f.y = vy*inv - py;
    ((float2*)out)[(size_t)b*HW_ + pix] = f;
}

// ---- launch -----------------------------------------------------------------
extern "C" void kernel_launch(void* const* d_in, const int* in_sizes, int n_in,
                              void* d_out, int out_size, void* d_ws, size_t ws_size,
                              hipStream_t stream) {
    // inputs: im(unused), mask, depth, T, Tc
    const float* mask  = (const float*)d_in[1];
    const float* depth = (const float*)d_in[2];
    const float* T     = (const float*)d_in[3];
    const float* Tc    = (const float*)d_in[4];
    float* out  = (float*)d_out;
    float* sums = (float*)d_ws;        // 256 floats
    float* cst  = sums + 256;          // 8 * 128 floats

    zero_sums  <<<1, 256, 0, stream>>>(sums);
    reduce_wmma<<<dim3(RBLK, B_), 256, 0, stream>>>(mask, depth, sums);
    make_consts<<<1, 64, 0, stream>>>(sums, T, Tc, cst);
    flow_main  <<<dim3(HW_/256, B_), 256, 0, stream>>>(mask, depth, cst, out);
}